// MGPATH_86414741995704
// MI455X (gfx1250) — compile-verified
//
#include <hip/hip_runtime.h>
#include <math.h>

typedef __attribute__((ext_vector_type(2))) float v2f;
typedef __attribute__((ext_vector_type(4))) float v4f;
typedef __attribute__((ext_vector_type(8))) float v8f;

#define M_ROWS   65536
#define D_DIM    1024
#define SK_BLOCKS 64
#define SK_TPB    256
#define ROWS_PT   (M_ROWS / (SK_BLOCKS * SK_TPB))   // 4 rows per thread

// ---------------------------------------------------------------------------
// Deterministic wave32 reduction
// ---------------------------------------------------------------------------
__device__ __forceinline__ float wave_sum(float x) {
#pragma unroll
  for (int off = 16; off > 0; off >>= 1) x += __shfl_xor(x, off, 32);
  return x;
}

// ---------------------------------------------------------------------------
// Hand-rolled grid-wide barrier (64 co-resident blocks), agent-scope atomics
// ---------------------------------------------------------------------------
__device__ __forceinline__ void grid_sync(int* arrive, int* gen) {
  __syncthreads();
  if (threadIdx.x == 0) {
    __threadfence();  // release our block's prior stores (agent scope)
    int g = __hip_atomic_load(gen, __ATOMIC_RELAXED, __HIP_MEMORY_SCOPE_AGENT);
    int prev = __hip_atomic_fetch_add(arrive, 1, __ATOMIC_ACQ_REL,
                                      __HIP_MEMORY_SCOPE_AGENT);
    if (prev == SK_BLOCKS - 1) {
      __hip_atomic_store(arrive, 0, __ATOMIC_RELAXED, __HIP_MEMORY_SCOPE_AGENT);
      __hip_atomic_store(gen, g + 1, __ATOMIC_RELEASE, __HIP_MEMORY_SCOPE_AGENT);
    } else {
      while (__hip_atomic_load(gen, __ATOMIC_RELAXED,
                               __HIP_MEMORY_SCOPE_AGENT) == g)
        __builtin_amdgcn_s_sleep(1);
    }
    __threadfence();  // acquire: invalidate WGP caches for whole block
  }
  __syncthreads();
}

// ---------------------------------------------------------------------------
// Kernel 0: reset barrier / flag state (graph replays must start clean)
// ---------------------------------------------------------------------------
__global__ void init_ctrl(float* ctrl) {
  int t = threadIdx.x;
  if (t < 8) ctrl[t] = 1.0f;                 // c0 = ones (hygiene)
  if (t >= 8 && t < 11) ((int*)ctrl)[t] = 0; // done, arrive, gen
}

// ---------------------------------------------------------------------------
// Kernel 1: sim[m][j] = dot(img[m], txt[j])  via V_WMMA_F32_16X16X4_F32
//   block = 256 thr (8 waves), wave handles a 16-row tile, grid = 512 blocks
// ---------------------------------------------------------------------------
__global__ __launch_bounds__(256) void sim_gemm(const float* __restrict__ img,
                                                const float* __restrict__ txt,
                                                float* __restrict__ simo) {
  __shared__ float sB[8 * 1032];  // 8 text vectors, padded stride (bank spread)
  const int tid = threadIdx.x;
#pragma unroll 4
  for (int idx = tid; idx < 8 * D_DIM; idx += 256) {
    int j = idx >> 10, k = idx & (D_DIM - 1);
    sB[j * 1032 + k] = txt[idx];
  }
  __syncthreads();

  const int wave = tid >> 5, lane = tid & 31;
  const int l = lane & 15, h = lane >> 4;
  const int rowBase = blockIdx.x * 128 + wave * 16;

  const float* Ap = img + (size_t)(rowBase + l) * D_DIM + 2 * h;
  const float* Bp = sB + (l & 7) * 1032 + 2 * h;  // lanes 8-15 duplicate (broadcast)

  v8f acc = {0.f, 0.f, 0.f, 0.f, 0.f, 0.f, 0.f, 0.f};
#pragma unroll 4
  for (int kk = 0; kk < D_DIM; kk += 4) {
    v2f a = *(const v2f*)(Ap + kk);   // A[M=l][kk+2h .. kk+2h+1]
    v2f b = *(const v2f*)(Bp + kk);   // B[kk+2h .. +1][N=l]
    acc = __builtin_amdgcn_wmma_f32_16x16x4_f32(false, a, false, b, (short)0,
                                                acc, false, false);
  }
  if (l < 8) {  // C layout: acc[v] = D[M = v + 8h][N = l]
#pragma unroll
    for (int v = 0; v < 8; ++v)
      simo[(size_t)(rowBase + v + 8 * h) * 8 + l] = acc[v];
  }
}

// ---------------------------------------------------------------------------
// Kernel 2: persistent Sinkhorn + final contraction.
//   K rows live in registers; per-iteration state is c[8]; grid barriers sync.
// ctrl layout (floats): [0..7] c, [8] done(int), [9] arrive(int), [10] gen(int),
//                       [16 + blk*12 + v] block partials
// ---------------------------------------------------------------------------
__global__ __launch_bounds__(SK_TPB) void sinkhorn_kernel(
    const float* __restrict__ simi, const float* __restrict__ logit_scale,
    float* ctrl, float* out) {
  float* c0g      = ctrl;
  int*   done_p   = (int*)(ctrl + 8);
  int*   arrive_p = (int*)(ctrl + 9);
  int*   gen_p    = (int*)(ctrl + 10);
  float* partials = ctrl + 16;

  const int tid  = threadIdx.x;
  const int wave = tid >> 5, lane = tid & 31;
  const int gtid = blockIdx.x * SK_TPB + tid;

  __shared__ float sRed[8 * 16];

  // Load my 4 rows of sim, build K = exp(10*(sim-1)); keep all in registers.
  float Kr[ROWS_PT][8], Sm[ROWS_PT][8], rprev[ROWS_PT][2];
#pragma unroll
  for (int i = 0; i < ROWS_PT; ++i) {
    int row = gtid + i * (SK_BLOCKS * SK_TPB);
    const v4f* p = (const v4f*)(simi + (size_t)row * 8);
    v4f lo = p[0], hi = p[1];
    Sm[i][0] = lo.x; Sm[i][1] = lo.y; Sm[i][2] = lo.z; Sm[i][3] = lo.w;
    Sm[i][4] = hi.x; Sm[i][5] = hi.y; Sm[i][6] = hi.z; Sm[i][7] = hi.w;
#pragma unroll
    for (int j = 0; j < 8; ++j) Kr[i][j] = __expf(10.f * (Sm[i][j] - 1.f));
    rprev[i][0] = 1.f; rprev[i][1] = 1.f;  // r0 = ones
  }
  float c0l[8];
#pragma unroll
  for (int j = 0; j < 8; ++j) c0l[j] = 1.f;  // c0 = ones

  const float invM = 1.0f / (float)M_ROWS;

  for (int it = 1; it <= 100; ++it) {
    float S[8] = {0.f, 0.f, 0.f, 0.f, 0.f, 0.f, 0.f, 0.f};
    float E = 0.f;
#pragma unroll
    for (int i = 0; i < ROWS_PT; ++i) {
      float d0 = 0.f, d1 = 0.f;
#pragma unroll
      for (int n = 0; n < 4; ++n) {
        d0 += Kr[i][2 * n]     * c0l[n];
        d1 += Kr[i][2 * n + 1] * c0l[4 + n];
      }
      float r0 = invM / d0, r1 = invM / d1;  // r = u / (K c)
      E += fabsf(r0 - rprev[i][0]) + fabsf(r1 - rprev[i][1]);
      rprev[i][0] = r0; rprev[i][1] = r1;
#pragma unroll
      for (int n = 0; n < 4; ++n) {
        S[n]     += Kr[i][2 * n]     * r0;   // K^T r
        S[4 + n] += Kr[i][2 * n + 1] * r1;
      }
    }
    // deterministic block reduction of 9 values
    float vals[9];
#pragma unroll
    for (int v = 0; v < 8; ++v) vals[v] = wave_sum(S[v]);
    vals[8] = wave_sum(E);
    if (lane == 0) {
#pragma unroll
      for (int v = 0; v < 9; ++v) sRed[wave * 16 + v] = vals[v];
    }
    __syncthreads();
    if (tid == 0) {
#pragma unroll
      for (int v = 0; v < 9; ++v) {
        float s = 0.f;
#pragma unroll
        for (int w = 0; w < 8; ++w) s += sRed[w * 16 + v];
        partials[blockIdx.x * 12 + v] = s;
      }
    }
    grid_sync(arrive_p, gen_p);

    if (blockIdx.x == 0) {  // deterministic 64-way reduce + c update
      float pv[9];
#pragma unroll
      for (int v = 0; v < 9; ++v)
        pv[v] = (tid < 64) ? partials[tid * 12 + v] : 0.f;
#pragma unroll
      for (int v = 0; v < 9; ++v) pv[v] = wave_sum(pv[v]);
      if (lane == 0 && wave < 2) {
#pragma unroll
        for (int v = 0; v < 9; ++v) sRed[wave * 16 + v] = pv[v];
      }
      __syncthreads();
      if (tid == 0) {
        float err = (sRed[8] + sRed[16 + 8]) * (1.0f / (2.0f * M_ROWS));
#pragma unroll
        for (int j = 0; j < 8; ++j)
          c0g[j] = 0.25f / (sRed[j] + sRed[16 + j]);   // c = v / (K^T r)
        *done_p = (err < 0.01f || it == 100) ? 1 : 0;
      }
    }
    grid_sync(arrive_p, gen_p);

    volatile float* vc = (volatile float*)c0g;
#pragma unroll
    for (int j = 0; j < 8; ++j) c0l[j] = vc[j];
    if (*(volatile int*)done_p) break;
  }

  // Final: out[b] = sum_m r_m,b * sum_n c[b][n]*K*sim
  float o0 = 0.f, o1 = 0.f;
#pragma unroll
  for (int i = 0; i < ROWS_PT; ++i) {
    float t0 = 0.f, t1 = 0.f;
#pragma unroll
    for (int n = 0; n < 4; ++n) {
      t0 += c0l[n]     * Kr[i][2 * n]     * Sm[i][2 * n];
      t1 += c0l[4 + n] * Kr[i][2 * n + 1] * Sm[i][2 * n + 1];
    }
    o0 += rprev[i][0] * t0;
    o1 += rprev[i][1] * t1;
  }
  o0 = wave_sum(o0); o1 = wave_sum(o1);
  if (lane == 0) { sRed[wave * 16 + 0] = o0; sRed[wave * 16 + 1] = o1; }
  __syncthreads();
  if (tid == 0) {
    float a = 0.f, b = 0.f;
#pragma unroll
    for (int w = 0; w < 8; ++w) { a += sRed[w * 16]; b += sRed[w * 16 + 1]; }
    partials[blockIdx.x * 12 + 0] = a;
    partials[blockIdx.x * 12 + 1] = b;
  }
  grid_sync(arrive_p, gen_p);
  if (blockIdx.x == 0) {
    float p0 = (tid < 64) ? partials[tid * 12 + 0] : 0.f;
    float p1 = (tid < 64) ? partials[tid * 12 + 1] : 0.f;
    p0 = wave_sum(p0); p1 = wave_sum(p1);
    if (lane == 0 && wave < 2) { sRed[wave * 16] = p0; sRed[wave * 16 + 1] = p1; }
    __syncthreads();
    if (tid == 0) {
      float sc = __expf(logit_scale[0]);
      out[0] = sc * (sRed[0] + sRed[16]);
      out[1] = sc * (sRed[1] + sRed[16 + 1]);
    }
  }
}

// ---------------------------------------------------------------------------
extern "C" void kernel_launch(void* const* d_in, const int* in_sizes, int n_in,
                              void* d_out, int out_size, void* d_ws,
                              size_t ws_size, hipStream_t stream) {
  const float* img = (const float*)d_in[0];   // [65536,1,1024] f32
  const float* txt = (const float*)d_in[1];   // [4,2,1024]     f32
  const float* ls  = (const float*)d_in[2];   // [1]            f32
  float* out = (float*)d_out;                 // 2 floats
  float* ws  = (float*)d_ws;
  float* simw = ws;                           // 65536*8 floats (2 MB)
  float* ctrl = ws + (size_t)M_ROWS * 8;      // barrier/c state + partials

  init_ctrl<<<1, 32, 0, stream>>>(ctrl);
  sim_gemm<<<M_ROWS / 128, 256, 0, stream>>>(img, txt, simw);
  sinkhorn_kernel<<<SK_BLOCKS, SK_TPB, 0, stream>>>(simw, ls, ctrl, out);
}